// PointsSampler_62835371540835
// MI455X (gfx1250) — compile-verified
//
#include <hip/hip_runtime.h>
#include <stdint.h>

// Problem constants (fixed by the reference).
#define BATCH   16
#define NPTS    131072
#define NSAMP   4096
#define WPB     8                    // workgroups cooperating per batch
#define TPB     1024                 // threads per workgroup (32 waves, wave32)
#define SLICE   (NPTS / WPB)         // 16384 points per WG
#define PPT     (SLICE / TPB)        // 16 points per thread (register-resident)
#define NWAVES  (TPB / 32)

// LDS carve-up (bytes): xyz slice (random access for winner lookup) | reduce | centroid
#define LDS_XYZ_BYTES   (SLICE * 3 * 4)            // 192 KiB
#define LDS_WRED_OFF    (LDS_XYZ_BYTES)
#define LDS_SC_OFF      (LDS_WRED_OFF + NWAVES * 8)
#define LDS_TOTAL_BYTES (LDS_SC_OFF + 16)          // ~192.3 KiB (< 320 KiB/WG)

// Scratch strides (pad hot sync lines to 64B).
#define CNT_STRIDE_U32  16
#define PUB_STRIDE_U64   8

typedef unsigned long long u64;

__device__ __forceinline__ u64 shfl_xor_u64(u64 v, int lanemask) {
    unsigned lo = (unsigned)v;
    unsigned hi = (unsigned)(v >> 32);
    lo = (unsigned)__shfl_xor((int)lo, lanemask, 32);
    hi = (unsigned)__shfl_xor((int)hi, lanemask, 32);
    return ((u64)hi << 32) | lo;
}

// CDNA5 async copy: 16B per lane, global -> LDS, tracked by ASYNCcnt.
__device__ __forceinline__ void async_g2lds_b128(unsigned lds_byte_off, u64 gaddr) {
    asm volatile("global_load_async_to_lds_b128 %0, %1, off"
                 :: "v"(lds_byte_off), "v"(gaddr) : "memory");
}
__device__ __forceinline__ void wait_asynccnt0() {
    asm volatile("s_wait_asynccnt 0" ::: "memory");
}

// Re-zero the inter-workgroup sync area every launch (graph-replay safe:
// counters are monotonic within a launch, zeroed between launches).
__global__ __launch_bounds__(512) void fps_init_kernel(uint32_t* sync_zero_base) {
    sync_zero_base[threadIdx.x] = 0u;   // 2 KiB = 512 dwords (cnt + epoch)
}

__global__ __launch_bounds__(TPB) void fps_main_kernel(
    const float* __restrict__ xyz,     // (B, N, 3) AoS
    int*         __restrict__ out,     // (B, NSAMP) int32 indices
    uint32_t*    __restrict__ cnt,     // [b*CNT_STRIDE_U32] monotonic arrivals
    uint32_t*    __restrict__ epoch,   // [b*CNT_STRIDE_U32] published round
    u64*         __restrict__ pubA,    // [b*PUB_STRIDE_U64] {cy,cx} bits
    u64*         __restrict__ pubB,    // [b*PUB_STRIDE_U64] {idx,cz} bits
    u64*         __restrict__ slots)   // [(b*WPB+w)*3] {key, {cy,cx}, {idx,cz}}
{
    extern __shared__ unsigned char smem[];
    float* lxyz  = (float*)smem;                          // [SLICE*3] AoS
    u64*   wred  = (u64*)  (smem + LDS_WRED_OFF);         // [NWAVES]
    float* scf   = (float*)(smem + LDS_SC_OFF);           // cx, cy, cz
    int*   sidxp = (int*)  (smem + LDS_SC_OFF + 12);

    const int b    = blockIdx.x / WPB;
    const int w    = blockIdx.x % WPB;
    const int t    = threadIdx.x;
    const int lane = t & 31;
    const int wid  = t >> 5;

    const float* __restrict__ P = xyz + (size_t)b * NPTS * 3;
    const int rbase = w * SLICE;

    // ---- one-time async stage of this WG's xyz slice into LDS (192 KiB) ----
    {
        const u64 gsrc = (u64)(uintptr_t)(P + (size_t)rbase * 3);   // 16B aligned
        const unsigned lds_base = (unsigned)(size_t)(void*)lxyz;
        #pragma unroll
        for (int j = 0; j < (LDS_XYZ_BYTES / 16) / TPB; ++j) {      // 12 chunks/thread
            const int c = t + (j << 10);
            async_g2lds_b128(lds_base + (unsigned)c * 16u, gsrc + (u64)c * 16u);
        }
    }

    if (t == 0) {               // first centroid is always global point 0
        scf[0] = P[0]; scf[1] = P[1]; scf[2] = P[2];
        *sidxp = 0;
    }
    wait_asynccnt0();
    __syncthreads();

    // ---- register-resident per-thread state: 16 points + running distances ----
    float px[PPT], py[PPT], pz[PPT], d[PPT];
    #pragma unroll
    for (int j = 0; j < PPT; ++j) {
        const int i = t + (j << 10);
        px[j] = lxyz[3 * i];
        py[j] = lxyz[3 * i + 1];
        pz[j] = lxyz[3 * i + 2];
        d[j]  = __builtin_inff();          // reference init
    }

    for (int s = 0; s < NSAMP; ++s) {
        const float cx = scf[0], cy = scf[1], cz = scf[2];
        if (w == 0 && t == 0) out[b * NSAMP + s] = *sidxp;   // emit carry-in

        // ---- pure-VALU min-update + in-lane argmax over 16 register points ----
        float bestD = -1.0f;    // all distances >= 0
        int   bestJ = 0;
        #pragma unroll
        for (int j = 0; j < PPT; ++j) {
            const float dx = px[j] - cx, dy = py[j] - cy, dz = pz[j] - cz;
            // mirror XLA: square elementwise, left-to-right sum, no fma contraction
            const float dsq = __fadd_rn(__fadd_rn(__fmul_rn(dx, dx),
                                                  __fmul_rn(dy, dy)),
                                        __fmul_rn(dz, dz));
            const float nd = fminf(d[j], dsq);
            d[j] = nd;
            if (nd > bestD) { bestD = nd; bestJ = j; }  // strict >, ascending j == ascending index
        }

        // tie-aware key over the GLOBAL index: bigger dist, then lower index
        const unsigned gi = (unsigned)(rbase + t + (bestJ << 10));
        u64 key = ((u64)__float_as_uint(bestD) << 32) | (unsigned)(~gi);

        #pragma unroll
        for (int off = 16; off > 0; off >>= 1) {        // wave32 xor-shuffle max
            u64 o = shfl_xor_u64(key, off);
            key = (o > key) ? o : key;
        }
        if (lane == 0) wred[wid] = key;
        __syncthreads();

        if (t == 0) {
            u64 k = wred[0];
            #pragma unroll
            for (int q = 1; q < NWAVES; ++q) { u64 o = wred[q]; k = (o > k) ? o : k; }
            const unsigned wgi = ~(unsigned)k;          // this WG's winner (global idx)
            const int li = (int)wgi - rbase;            // local index into LDS copy

            // candidate coords from LDS (keeps winner's centroid off the critical path)
            const float qx = lxyz[3 * li], qy = lxyz[3 * li + 1], qz = lxyz[3 * li + 2];

            u64* sl = slots + (size_t)(b * WPB + w) * 3;
            __hip_atomic_store(&sl[0], k, __ATOMIC_RELAXED, __HIP_MEMORY_SCOPE_AGENT);
            __hip_atomic_store(&sl[1],
                ((u64)__float_as_uint(qy) << 32) | __float_as_uint(qx),
                __ATOMIC_RELAXED, __HIP_MEMORY_SCOPE_AGENT);
            __hip_atomic_store(&sl[2],
                ((u64)wgi << 32) | __float_as_uint(qz),
                __ATOMIC_RELAXED, __HIP_MEMORY_SCOPE_AGENT);

            const uint32_t old = __hip_atomic_fetch_add(
                &cnt[b * CNT_STRIDE_U32], 1u, __ATOMIC_ACQ_REL, __HIP_MEMORY_SCOPE_AGENT);

            if (old + 1u == (uint32_t)WPB * (uint32_t)(s + 1)) {
                // last arriver of this round: reduce the WPB slot keys
                u64 bk = 0; int bw = 0;
                #pragma unroll
                for (int q = 0; q < WPB; ++q) {
                    u64 kk = __hip_atomic_load(&slots[(size_t)(b * WPB + q) * 3],
                                               __ATOMIC_RELAXED, __HIP_MEMORY_SCOPE_AGENT);
                    if (kk > bk) { bk = kk; bw = q; }
                }
                const u64 a = __hip_atomic_load(&slots[(size_t)(b * WPB + bw) * 3 + 1],
                                                __ATOMIC_RELAXED, __HIP_MEMORY_SCOPE_AGENT);
                const u64 v = __hip_atomic_load(&slots[(size_t)(b * WPB + bw) * 3 + 2],
                                                __ATOMIC_RELAXED, __HIP_MEMORY_SCOPE_AGENT);
                __hip_atomic_store(&pubA[b * PUB_STRIDE_U64], a,
                                   __ATOMIC_RELAXED, __HIP_MEMORY_SCOPE_AGENT);
                __hip_atomic_store(&pubB[b * PUB_STRIDE_U64], v,
                                   __ATOMIC_RELAXED, __HIP_MEMORY_SCOPE_AGENT);
                __hip_atomic_store(&epoch[b * CNT_STRIDE_U32], (uint32_t)(s + 1),
                                   __ATOMIC_RELEASE, __HIP_MEMORY_SCOPE_AGENT);
            }

            // spin (release->acquire pairing; s_sleep throttles L2 polling)
            while (__hip_atomic_load(&epoch[b * CNT_STRIDE_U32],
                                     __ATOMIC_ACQUIRE, __HIP_MEMORY_SCOPE_AGENT)
                   < (uint32_t)(s + 1)) {
                __builtin_amdgcn_s_sleep(1);
            }
            const u64 a = __hip_atomic_load(&pubA[b * PUB_STRIDE_U64],
                                            __ATOMIC_RELAXED, __HIP_MEMORY_SCOPE_AGENT);
            const u64 v = __hip_atomic_load(&pubB[b * PUB_STRIDE_U64],
                                            __ATOMIC_RELAXED, __HIP_MEMORY_SCOPE_AGENT);
            scf[0] = __uint_as_float((unsigned)a);
            scf[1] = __uint_as_float((unsigned)(a >> 32));
            scf[2] = __uint_as_float((unsigned)v);
            *sidxp = (int)(v >> 32);
        }
        __syncthreads();   // new centroid visible; old sc was read before wred barrier
    }
}

extern "C" void kernel_launch(void* const* d_in, const int* in_sizes, int n_in,
                              void* d_out, int out_size, void* d_ws, size_t ws_size,
                              hipStream_t stream) {
    (void)in_sizes; (void)n_in; (void)out_size; (void)ws_size;
    const float* xyz = (const float*)d_in[0];
    int* out = (int*)d_out;   // reference output dtype is int32 indices

    // Scratch: only the ~8 KiB sync block (all bulk state lives in LDS/VGPRs).
    uint8_t* p = (uint8_t*)d_ws;
    uint32_t* cnt   = (uint32_t*)p;  p += BATCH * CNT_STRIDE_U32 * sizeof(uint32_t); // 1 KiB
    uint32_t* epoch = (uint32_t*)p;  p += BATCH * CNT_STRIDE_U32 * sizeof(uint32_t); // 1 KiB
    u64* pubA  = (u64*)p;            p += BATCH * PUB_STRIDE_U64 * sizeof(u64);      // 1 KiB
    u64* pubB  = (u64*)p;            p += BATCH * PUB_STRIDE_U64 * sizeof(u64);      // 1 KiB
    u64* slots = (u64*)p;            // BATCH*WPB*3*8 = 3 KiB

    fps_init_kernel<<<dim3(1), dim3(512), 0, stream>>>(cnt);
    fps_main_kernel<<<dim3(BATCH * WPB), dim3(TPB), LDS_TOTAL_BYTES, stream>>>(
        xyz, out, cnt, epoch, pubA, pubB, slots);
}